// RoIAlign_60507499266507
// MI455X (gfx1250) — compile-verified
//
#include <hip/hip_runtime.h>
#include <stdint.h>

// ---- problem constants (fixed by the reference's setup_inputs) -------------
#define FH 200
#define FW 320
#define RS 14
#define NBINS (RS * RS)   // 196

// ---- feature availability guards ------------------------------------------
#if defined(__has_builtin)
#if __has_builtin(__builtin_amdgcn_global_load_async_to_lds_b32)
#define USE_ASYNC_LDS 1
#endif
#if __has_builtin(__builtin_amdgcn_wmma_f32_16x16x4_f32)
#define USE_WMMA 1
#endif
#endif
#ifndef USE_ASYNC_LDS
#define USE_ASYNC_LDS 0
#endif
#ifndef USE_WMMA
#define USE_WMMA 0
#endif

typedef float v2f __attribute__((ext_vector_type(2)));
typedef float v8f __attribute__((ext_vector_type(8)));

// Order-preserving float <-> uint encoding (monotonic for all non-NaN floats)
// so we can use integer max atomics (ds_max_u32 / global_atomic_max_u32).
__device__ __forceinline__ unsigned encf(float f) {
  unsigned u = __float_as_uint(f);
  return (u & 0x80000000u) ? ~u : (u | 0x80000000u);
}
__device__ __forceinline__ float decf(unsigned u) {
  unsigned b = (u & 0x80000000u) ? (u & 0x7fffffffu) : ~u;
  return __uint_as_float(b);
}

// ---------------------------------------------------------------------------
// Phase 0: init the 196-entry global bin accumulator (0u, below every encoded non-NaN float).
// ---------------------------------------------------------------------------
__global__ void roialign_init_bins(unsigned* __restrict__ gbins) {
  if (threadIdx.x < NBINS) gbins[threadIdx.x] = 0u;
}

// ---------------------------------------------------------------------------
// Phase 1: per-ROI gather + max reduction.
// grid = R blocks, block = 256 threads (thread == channel).
// Coordinates are block-uniform per bin iteration -> SALU; per-lane vmem
// loads differ only by the channel-plane base. Feature (65.5 MB) is L2
// resident on MI455X (192 MB L2), so this is an L2-bound gather.
// ---------------------------------------------------------------------------
__global__ __launch_bounds__(256) void roialign_max_kernel(
    const float* __restrict__ feat, const float* __restrict__ rois,
    unsigned* __restrict__ gbins) {
  const int r = blockIdx.x;
  const int c = threadIdx.x;
  const int lane = threadIdx.x & 31;

  __shared__ float s_roi[4];
  __shared__ unsigned s_bins[NBINS];

  for (int i = threadIdx.x; i < NBINS; i += 256) s_bins[i] = 0u;

#if USE_ASYNC_LDS
  if (threadIdx.x < 4) {
    // Builtin prototype (per hipcc diagnostic) takes generic int* pointers;
    // the backend infers global/LDS address spaces.
    int* gptr = (int*)(uintptr_t)(rois + (size_t)r * 4 + threadIdx.x);
    int* lptr = (int*)&s_roi[threadIdx.x];
    __builtin_amdgcn_global_load_async_to_lds_b32(gptr, lptr, 0, 0);
  }
  asm volatile("s_wait_asynccnt 0x0" ::: "memory");
#else
  if (threadIdx.x < 4) s_roi[threadIdx.x] = rois[(size_t)r * 4 + threadIdx.x];
#endif
  __syncthreads();

  const float y0 = s_roi[0];
  const float x0 = s_roi[1];
  const float sh = (s_roi[2] - y0) * (1.0f / (float)RS);
  const float sw = (s_roi[3] - x0) * (1.0f / (float)RS);
  const float f13 = 1.0f / 3.0f, f23 = 2.0f / 3.0f;

  const float* plane = feat + (size_t)c * (FH * FW);

  for (int m = 0; m < RS; ++m) {
    const float ybase = y0 + sh * (float)m;
    for (int n = 0; n < RS; ++n) {
      const float xbase = x0 + sw * (float)n;
      float v = -3.402823466e38f;
#pragma unroll
      for (int sy = 0; sy < 2; ++sy) {
        const float y = ybase + sh * (sy ? f23 : f13);
        const float fy = floorf(y);
        const int y1i = min(max((int)fy, 0), FH - 1);
        const int y2i = min(max((int)fy + 1, 0), FH - 1);
        // weights from CLAMPED indices (matches reference exactly)
        const float wy_lo = y - (float)y1i;
        const float wy_hi = (float)y2i - y;
        const float* row1 = plane + y1i * FW;
        const float* row2 = plane + y2i * FW;
#pragma unroll
        for (int sx = 0; sx < 2; ++sx) {
          const float x = xbase + sw * (sx ? f23 : f13);
          const float fx = floorf(x);
          const int x1i = min(max((int)fx, 0), FW - 1);
          const int x2i = min(max((int)fx + 1, 0), FW - 1);
          const float wx_lo = x - (float)x1i;
          const float wx_hi = (float)x2i - x;
          const float p = row1[x1i] * wx_hi + row1[x2i] * wx_lo;
          const float q = row2[x1i] * wx_hi + row2[x2i] * wx_lo;
          v = fmaxf(v, p * wy_hi + q * wy_lo);
        }
      }
      // wave32 max tree, then one ds_max_u32 per wave per bin
      for (int off = 16; off > 0; off >>= 1)
        v = fmaxf(v, __shfl_xor(v, off, 32));
      if (lane == 0) atomicMax(&s_bins[m * RS + n], encf(v));
    }
  }
  __syncthreads();
  for (int i = threadIdx.x; i < NBINS; i += 256)
    atomicMax(&gbins[i], s_bins[i]);
}

// ---------------------------------------------------------------------------
// Phase 2: broadcast bin_max to [R*C, 196] through the matrix pipe.
// D = A x B with A = e0 row-selector (exact 1.0/0.0) and B row 0 = bin
// values gives D[i][j] = bin[p0+j] bit-exactly in f32. One 16x16 tile per
// wave; 196 columns = 12 full tiles + one overlapping tail tile (idempotent
// duplicate writes). Store-bound: 25.7 MB -> ~1.1 us at 23.3 TB/s.
// ---------------------------------------------------------------------------
__global__ __launch_bounds__(256) void roialign_bcast_kernel(
    const unsigned* __restrict__ gbins, float* __restrict__ out,
    int rowTiles) {
  const int wave = blockIdx.x * (blockDim.x >> 5) + (threadIdx.x >> 5);
  const int lane = threadIdx.x & 31;
  const int nTiles = rowTiles * 13;
  if (wave >= nTiles) return;  // wave-uniform branch: EXEC stays all-ones

  const int colTile = wave % 13;
  const int rowTile = wave / 13;
  const int p0 = (colTile < 12) ? colTile * 16 : (NBINS - 16);
  const int rc0 = rowTile * 16;

#if USE_WMMA
  v2f a, b;
  a.x = (lane < 16) ? 1.0f : 0.0f;  // A[M][K]: K=0 column = 1, K=1..3 = 0
  a.y = 0.0f;
  float bv = (lane < 16) ? decf(gbins[p0 + lane]) : 0.0f;
  b.x = bv;                         // B[K][N]: K=0 row = bins, K=2 row = 0
  b.y = 0.0f;                       // K=1, K=3 rows = 0
  v8f cacc = {};
  v8f d = __builtin_amdgcn_wmma_f32_16x16x4_f32(
      false, a, false, b, (short)0, cacc, false, false);
  const int col = p0 + (lane & 15);
  const int rbase = rc0 + ((lane >> 4) << 3);  // lanes 16..31 hold M+8
#pragma unroll
  for (int k = 0; k < 8; ++k)
    out[(size_t)(rbase + k) * NBINS + col] = d[k];
#else
  const int col = p0 + (lane & 15);
  const float bv = decf(gbins[col]);
  const int rbase = rc0 + ((lane >> 4) << 3);
#pragma unroll
  for (int k = 0; k < 8; ++k)
    out[(size_t)(rbase + k) * NBINS + col] = bv;
#endif
}

// ---------------------------------------------------------------------------
extern "C" void kernel_launch(void* const* d_in, const int* in_sizes, int n_in,
                              void* d_out, int out_size, void* d_ws,
                              size_t ws_size, hipStream_t stream) {
  const float* feat = (const float*)d_in[0];   // [1,256,200,320] f32
  const float* rois = (const float*)d_in[1];   // [128,4] f32
  float* out = (float*)d_out;                  // [128,256,14,14] f32
  unsigned* gbins = (unsigned*)d_ws;           // 196 encoded-max accumulators

  const int R = in_sizes[1] / 4;               // 128
  const int C = in_sizes[0] / (FH * FW);       // 256
  const int RC = R * C;                        // 32768 (multiple of 16)
  const int rowTiles = RC / 16;                // 2048
  const int nTiles = rowTiles * 13;            // 26624
  const int bblocks = (nTiles + 7) / 8;        // 8 waves per 256-thread block

  roialign_init_bins<<<1, 256, 0, stream>>>(gbins);
  roialign_max_kernel<<<R, 256, 0, stream>>>(feat, rois, gbins);
  roialign_bcast_kernel<<<bblocks, 256, 0, stream>>>(gbins, out, rowTiles);
}